// HT2IM_77163382440036
// MI455X (gfx1250) — compile-verified
//
#include <hip/hip_runtime.h>
#include <stdint.h>

// ---------------- problem constants (from reference) ----------------
#define IM_PIX    16384        // 128*128
#define HT_PIX    10980        // 183*60
#define BC        256          // B*C channels
#define CH_PER_WG 4            // channels privatized per workgroup
#define NUM_WG    (BC / CH_PER_WG)   // 64 workgroups, exclusive channel ownership
#define THREADS   1024         // 32 waves (wave32)
#define CHUNK     1024         // votes staged per buffer slot
#define STAGE_FLTS (3 * CHUNK) // ht/im/w per slot
// LDS: 4ch*16384px*4B = 256KB image + 3 triple-buffer slots * 12KB = 292KB < 320KB
#define SMEM_FLTS (IM_PIX * CH_PER_WG + 3 * STAGE_FLTS)

// ---------------- gfx1250 async global->LDS path (guarded) ----------------
#if defined(__has_builtin)
#if __has_builtin(__builtin_amdgcn_global_load_async_to_lds_b128)
#define USE_ASYNC_LDS 1
#endif
#if __has_builtin(__builtin_amdgcn_s_wait_asynccnt)
#define HAVE_WAIT_ASYNC 1
#endif
#endif

// Builtin signature (verified via compiler diagnostic):
//   (v4i AS(1)* gsrc, v4i AS(3)* ldst, imm offset, imm cpol)
typedef int v4i __attribute__((vector_size(16)));
typedef __attribute__((address_space(1))) v4i gv4i_t;
typedef __attribute__((address_space(3))) v4i lv4i_t;

template <int N>
__device__ __forceinline__ void wait_async() {
#if defined(HAVE_WAIT_ASYNC)
  __builtin_amdgcn_s_wait_asynccnt(N);
#elif defined(USE_ASYNC_LDS)
  if constexpr (N == 0) { asm volatile("s_wait_asynccnt 0x0" ::: "memory"); }
  else                  { asm volatile("s_wait_asynccnt 0x1" ::: "memory"); }
#endif
}

// Stage CHUNK votes' (ht_index, im_index, weight) into an LDS slot.
// Waves 0..23 each issue ONE b128 async op (32 lanes x 16B = 512B),
// covering 3 arrays x 4KB; per-wave ASYNCcnt tracks it for pipelining.
__device__ __forceinline__ void issue_chunk(const int* __restrict__ ht,
                                            const int* __restrict__ im,
                                            const float* __restrict__ wt,
                                            int vb, float* buf,
                                            int wave, int lane, int tid) {
#if defined(USE_ASYNC_LDS)
  if (wave < 24) {
    const int arr  = wave >> 3;                       // 0:ht 1:im 2:wt
    const int eoff = ((wave & 7) << 7) + (lane << 2); // element offset 0..1023
    const void* g = (arr == 0) ? (const void*)(ht + vb + eoff)
                  : (arr == 1) ? (const void*)(im + vb + eoff)
                               : (const void*)(wt + vb + eoff);
    float* l = buf + arr * CHUNK + eoff;
    __builtin_amdgcn_global_load_async_to_lds_b128((gv4i_t*)g, (lv4i_t*)l,
                                                   /*offset=*/0, /*cpol=*/0);
  }
#else
  // Fallback: synchronous cooperative copy through LDS.
  for (int i = tid; i < CHUNK; i += THREADS) {
    ((int*)buf)[i]         = ht[vb + i];
    ((int*)buf)[CHUNK + i] = im[vb + i];
    buf[2 * CHUNK + i]     = wt[vb + i];
  }
  (void)wave; (void)lane;
#endif
}

__global__ void __launch_bounds__(THREADS, 1)
HT2IM_77163382440036_kernel(const float* __restrict__ input_ht,
                            const int*   __restrict__ ht_index,
                            const int*   __restrict__ im_index,
                            const float* __restrict__ weight,
                            float* __restrict__ out, int M) {
  extern __shared__ __align__(16) float smem[];
  float* img = smem;                              // [IM_PIX][CH_PER_WG]

  const int tid  = threadIdx.x;
  const int wave = tid >> 5;
  const int lane = tid & 31;
  const int cg   = blockIdx.x;                    // channel group (exclusive owner)
  const int ch0  = cg * CH_PER_WG;
  const int c    = tid & (CH_PER_WG - 1);         // channel within group
  const int vsub = tid >> 2;                      // 0..255: vote slot within pass

  // ---- zero private output image in LDS (16 float4 stores per thread) ----
  float4* img4 = (float4*)img;
#pragma unroll
  for (int i = 0; i < (IM_PIX * CH_PER_WG / 4) / THREADS; ++i)
    img4[tid + i * THREADS] = make_float4(0.f, 0.f, 0.f, 0.f);

  const float* xbase = input_ht + (size_t)ch0 * HT_PIX;  // this WG's 4 channel rows

  const int nchunks = M / CHUNK;

  // Triple-buffered stage slots: reads of a slot at iter j and its re-fill
  // issue at iter j+2 are separated by the completed barrier at j+1, so a
  // single barrier per iteration is sufficient.
  float* b0 = smem + IM_PIX * CH_PER_WG;
  float* b1 = b0 + STAGE_FLTS;
  float* b2 = b1 + STAGE_FLTS;

  __syncthreads();   // zero-init visible before any ds_add

  if (nchunks > 0) {
    // prologue: stage chunk 0 into slot b0
    issue_chunk(ht_index, im_index, weight, 0, b0, wave, lane, tid);

    for (int k = 0; k < nchunks; ++k) {
      if (k + 1 < nchunks) {
        issue_chunk(ht_index, im_index, weight, (k + 1) * CHUNK, b1, wave, lane, tid);
        wait_async<1>();   // per-wave: chunk k's single op complete
      } else {
        wait_async<0>();   // drain last chunk
      }
      __syncthreads();     // chunk k visible; all waves done with slot b1's old data

      // prefetch vote stream ahead into L2 (global_prefetch_b8)
      if (tid == 0 && (k + 4) < nchunks) {
        __builtin_prefetch(ht_index + (size_t)(k + 4) * CHUNK, 0, 3);
        __builtin_prefetch(im_index + (size_t)(k + 4) * CHUNK, 0, 3);
        __builtin_prefetch(weight   + (size_t)(k + 4) * CHUNK, 0, 3);
      }

      // ---- process chunk k from b0: 4 passes of (256 votes x 4 channels) ----
      float* cur = b0;
#pragma unroll
      for (int r = 0; r < CHUNK / 256; ++r) {
        const int vi  = (r << 8) + vsub;
        const int hti = ((const int*)cur)[vi];
        const int imi = ((const int*)cur)[CHUNK + vi];
        const float w = cur[2 * CHUNK + vi];
        const float x = xbase[(size_t)c * HT_PIX + hti];
        atomicAdd(&img[(imi << 2) + c], x * w);   // ds_add_f32, WG-private
      }

      // rotate slots: (b0,b1,b2) <- (b1,b2,b0)
      float* t = b0; b0 = b1; b1 = b2; b2 = t;
    }
  }

  // ---- tail votes (M % CHUNK), direct global loads ----
  const int vb   = nchunks * CHUNK;
  const int tail = M - vb;
  for (int t = vsub; t < tail; t += 256) {
    const int hti = ht_index[vb + t];
    const int imi = im_index[vb + t];
    const float w = weight[vb + t];
    const float x = xbase[(size_t)c * HT_PIX + hti];
    atomicAdd(&img[(imi << 2) + c], x * w);
  }
  __syncthreads();

  // ---- flush: exclusive ownership -> plain coalesced stores, no atomics ----
#pragma unroll
  for (int cc = 0; cc < CH_PER_WG; ++cc) {
    float* orow = out + (size_t)(ch0 + cc) * IM_PIX;
    for (int px = tid; px < IM_PIX; px += THREADS)
      orow[px] = img[(px << 2) + cc];
  }
}

extern "C" void kernel_launch(void* const* d_in, const int* in_sizes, int n_in,
                              void* d_out, int out_size, void* d_ws, size_t ws_size,
                              hipStream_t stream) {
  const float* input_ht = (const float*)d_in[0];
  const int*   ht_index = (const int*)d_in[1];
  const int*   im_index = (const int*)d_in[2];
  const float* weight   = (const float*)d_in[3];
  const int    M        = in_sizes[1];

  const size_t shmem = (size_t)SMEM_FLTS * sizeof(float);  // 299,008 B < 320 KB
  HT2IM_77163382440036_kernel<<<NUM_WG, THREADS, shmem, stream>>>(
      input_ht, ht_index, im_index, weight, (float*)d_out, M);
  (void)n_in; (void)out_size; (void)d_ws; (void)ws_size;
}